// GRGNCell_66975720014311
// MI455X (gfx1250) — compile-verified
//
#include <hip/hip_runtime.h>
#include <hip/hip_bf16.h>
#include <math.h>

// ---- Problem constants (from reference) ----
#define B_      4
#define T_      12
#define N_      3000
#define E_      96000
#define H_      64
#define MMIX_   32
#define GMM_    96          // (F_IN+2)*M
#define RNNIN_  161         // F_IN + GMM + H
#define DIFF_   675         // 3 * (RNNIN_ + H_)
#define FRONT_  225         // RNNIN_ + H_
#define R_      (B_*N_)     // 12000 rows (multiple of 16 -> no row guards)
#define ZLD_    164         // zin leading dim (161 padded to mult of 4)
#define DLD_    676         // diffin leading dim (675 padded to mult of 4)
#define XLD_    164         // xhat2 leading dim (161 padded to mult of 4)

typedef __attribute__((ext_vector_type(2))) float v2f;
typedef __attribute__((ext_vector_type(8))) float v8f;

enum { ACT_NONE = 0, ACT_SIGMOID = 1, ACT_TANH = 2 };

// =====================================================================
// Weight pre-pack into WMMA B-fragment order, zero-padded:
//   Wp[((ks*2 + hf)*Npad + c)*2 + j] = W[(4*ks + 2*hf + j), c]
// so each lane's (K, K+1) pair is one contiguous b64, coalesced across lanes.
// Supports fusing two weight matrices side-by-side ([W0 | W1]).
// =====================================================================
__global__ void pack_w_dual(const float* __restrict__ W0, int n0,
                            const float* __restrict__ W1, int n1,
                            float* __restrict__ Wp, int K, int Kpad, int Npad)
{
    long i = (long)blockIdx.x * blockDim.x + threadIdx.x;
    if (i >= (long)Kpad * Npad) return;
    const int  j  = (int)(i & 1);
    const long p  = i >> 1;
    const int  c  = (int)(p % Npad);
    const long tq = p / Npad;
    const int  hf = (int)(tq & 1);
    const int  ks = (int)(tq >> 1);
    const int  kk = ks * 4 + hf * 2 + j;
    float v = 0.0f;
    if (kk < K) {
        if (c < n0)            v = W0[(long)kk * n0 + c];
        else if (c < n0 + n1)  v = W1[(long)kk * n1 + (c - n0)];
    }
    Wp[i] = v;
}

__global__ void pack_bias_dual(const float* __restrict__ b0, int n0,
                               const float* __restrict__ b1, int n1,
                               float* __restrict__ outb)
{
    int i = blockIdx.x * blockDim.x + threadIdx.x;
    if (i >= n0 + n1) return;
    outb[i] = (i < n0) ? b0[i] : b1[i - n0];
}

// =====================================================================
// GEMM: C = act(A @ W + bias). One wave32 computes a 16x32 tile with two
// V_WMMA_F32_16X16X4_F32 accumulator chains. Branch-free, vector-b64 K loop.
//   A 16x4 frag: lanes 0-15 -> K={k,k+1}, lanes 16-31 -> K={k+2,k+3}
//   B 4x16 frag: same K split, N = lane&15 (from packed Wp)
// =====================================================================
__global__ void gemm_wmma_nt2(const float* __restrict__ A, int lda,
                              const float* __restrict__ Wp, int Npad,
                              const float* __restrict__ bias,
                              float* __restrict__ C, int ldc,
                              int tilesM, int groupsN, int Kpad, int Ncols, int act)
{
    const int lane = threadIdx.x & 31;
    const int wave = threadIdx.x >> 5;
    const long gtile = (long)blockIdx.x * (blockDim.x >> 5) + wave;
    if (gtile >= (long)tilesM * groupsN) return;      // wave-uniform exit
    const int tm = (int)(gtile % tilesM);
    const int tg = (int)(gtile / tilesM);
    const int row0 = tm << 4;
    const int colg = tg << 5;

    const int hf  = lane >> 4;       // half-wave id -> K offset 0 / 2
    const int l16 = lane & 15;

    const float* __restrict__ Arow = A + (long)(row0 + l16) * lda + (hf << 1);
    const float* __restrict__ Wb   = Wp + ((long)hf * Npad + colg + l16) * 2;
    const long wstep = (long)4 * Npad;               // floats per k-group

    v8f acc0 = {}, acc1 = {};
#pragma unroll 4
    for (int k = 0; k < Kpad; k += 4) {
        const v2f a = *(const v2f*)(Arow + k);
        const float* wp = Wb + (long)(k >> 2) * wstep;
        const v2f b0 = *(const v2f*)(wp);
        const v2f b1 = *(const v2f*)(wp + 32);
        acc0 = __builtin_amdgcn_wmma_f32_16x16x4_f32(false, a, false, b0, (short)0, acc0, false, false);
        acc1 = __builtin_amdgcn_wmma_f32_16x16x4_f32(false, a, false, b1, (short)0, acc1, false, false);
    }

#pragma unroll
    for (int nt = 0; nt < 2; ++nt) {
        const int bcol = colg + (nt << 4) + l16;
        if (bcol >= Ncols) continue;                 // epilogue-only guard
        const float bv = bias ? bias[bcol] : 0.0f;
        const v8f acc = nt ? acc1 : acc0;
#pragma unroll
        for (int i = 0; i < 8; ++i) {
            const int r = row0 + i + (hf << 3);
            float v = acc[i] + bv;
            if (act == ACT_SIGMOID)   v = 1.0f / (1.0f + __expf(-v));
            else if (act == ACT_TANH) v = tanhf(v);
            C[(long)r * ldc + bcol] = v;
        }
    }
}

// ---- zero fill (flat) ----
__global__ void fill0(float* __restrict__ p, long n)
{
    long i = (long)blockIdx.x * blockDim.x + threadIdx.x;
    if (i < n) p[i] = 0.0f;
}

// ---- zero fill (strided sub-block: rows x width, leading dim ld) ----
__global__ void fill0_2d(float* __restrict__ p, int rows, int width, int ld)
{
    long i = (long)blockIdx.x * blockDim.x + threadIdx.x;
    if (i >= (long)rows * width) return;
    int r = (int)(i / width), c = (int)(i % width);
    p[(long)r * ld + c] = 0.0f;
}

// ---- edge aggregation: out[b,dst,c] += in[b,src,c] * w[e] (hw fp32 atomics) ----
__global__ void agg_edges(const float* __restrict__ in, float* __restrict__ outp,
                          const int* __restrict__ src, const int* __restrict__ dst,
                          const float* __restrict__ w, int width, int ld)
{
    long t = (long)blockIdx.x * blockDim.x + threadIdx.x;
    const long total = (long)B_ * E_ * width;
    if (t >= total) return;
    const int c = (int)(t % width);
    const long eb = t / width;
    const int e = (int)(eb % E_);
    const int b = (int)(eb / E_);
    const float val = in[((long)b * N_ + src[e]) * ld + c] * w[e];
    unsafeAtomicAdd(&outp[((long)b * N_ + dst[e]) * ld + c], val);
}

// ---- GMM head: s1[r,0:32]=mu, [32:64]=softplus, [64:96]=softmax -> predictions ----
__global__ void gmm_head(float* __restrict__ s1, float* __restrict__ pred, int t)
{
    const int lane = threadIdx.x & 31;
    const int wave = threadIdx.x >> 5;
    const long row = (long)blockIdx.x * (blockDim.x >> 5) + wave;
    if (row >= (long)R_) return;                     // wave-uniform
    float* p = s1 + row * GMM_;
    float sv = p[MMIX_ + lane];
    sv = (sv > 20.0f) ? sv : log1pf(__expf(sv));
    float pv = p[2 * MMIX_ + lane];
    float mx = pv;
    for (int o = 16; o >= 1; o >>= 1) mx = fmaxf(mx, __shfl_xor(mx, o, 32));
    float ex = __expf(pv - mx);
    float sum = ex;
    for (int o = 16; o >= 1; o >>= 1) sum += __shfl_xor(sum, o, 32);
    const float pi = ex / sum;
    p[MMIX_ + lane] = sv;
    p[2 * MMIX_ + lane] = pi;
    const int b = (int)(row / N_), n = (int)(row % N_);
    float* o = pred + (((long)b * T_ + t) * N_ + n) * GMM_;
    o[lane] = p[lane];
    o[MMIX_ + lane] = sv;
    o[2 * MMIX_ + lane] = pi;
}

// ---- zin = [x_t | xs_hat_1 | h] (161 wide, ld ZLD_) ----
__global__ void assemble_zin(const float* __restrict__ x, const float* __restrict__ s1,
                             const float* __restrict__ h, float* __restrict__ zin, int t)
{
    long i = (long)blockIdx.x * blockDim.x + threadIdx.x;
    if (i >= (long)R_ * RNNIN_) return;
    const int r = (int)(i / RNNIN_), c = (int)(i % RNNIN_);
    const int b = r / N_, n = r % N_;
    float v;
    if (c == 0)              v = x[((long)b * T_ + t) * N_ + n];
    else if (c < 1 + GMM_)   v = s1[(long)r * GMM_ + (c - 1)];
    else                     v = h[(long)r * H_ + (c - 1 - GMM_)];
    zin[(long)r * ZLD_ + c] = v;
}

// ---- repr = relu([conv | h]) (128 wide) + representations out ----
__global__ void assemble_repr(const float* __restrict__ conv, const float* __restrict__ h,
                              float* __restrict__ reprb, float* __restrict__ reps, int t)
{
    long i = (long)blockIdx.x * blockDim.x + threadIdx.x;
    if (i >= (long)R_ * 2 * H_) return;
    const int r = (int)(i / (2 * H_)), c = (int)(i % (2 * H_));
    float v = (c < H_) ? conv[(long)r * H_ + c] : h[(long)r * H_ + (c - H_)];
    v = fmaxf(v, 0.0f);
    reprb[i] = v;
    const int b = r / N_, n = r % N_;
    reps[(((long)b * T_ + t) * N_ + n) * (2 * H_) + c] = v;
}

// ---- copy xs_hat_2 (ld XLD_) to generations output ----
__global__ void copy_gen(const float* __restrict__ xhat2, float* __restrict__ gen, int t)
{
    long i = (long)blockIdx.x * blockDim.x + threadIdx.x;
    if (i >= (long)R_ * RNNIN_) return;
    const int r = (int)(i / RNNIN_), c = (int)(i % RNNIN_);
    const int b = r / N_, n = r % N_;
    gen[(((long)b * T_ + t) * N_ + n) * RNNIN_ + c] = xhat2[(long)r * XLD_ + c];
}

// ---- diffin[:,0:225] = [xs_hat_2 | (use_r ? r*h : h)], r from rg[:,0:64] ----
__global__ void assemble_front(const float* __restrict__ xhat2, const float* __restrict__ h,
                               const float* __restrict__ rg, float* __restrict__ diffin,
                               int use_r)
{
    long i = (long)blockIdx.x * blockDim.x + threadIdx.x;
    if (i >= (long)R_ * FRONT_) return;
    const int r = (int)(i / FRONT_), c = (int)(i % FRONT_);
    float v;
    if (c < RNNIN_) v = xhat2[(long)r * XLD_ + c];
    else {
        const int cc = c - RNNIN_;
        const float hv = h[(long)r * H_ + cc];
        v = use_r ? rg[(long)r * (2 * H_) + cc] * hv : hv;
    }
    diffin[(long)r * DLD_ + c] = v;
}

// ---- h = g*h + (1-g)*c, g from rg[:,64:128]; write states out ----
__global__ void h_update(float* __restrict__ h, const float* __restrict__ rg,
                         const float* __restrict__ cbuf, float* __restrict__ states, int t)
{
    long i = (long)blockIdx.x * blockDim.x + threadIdx.x;
    if (i >= (long)R_ * H_) return;
    const int r = (int)(i / H_), c = (int)(i % H_);
    const float g = rg[(long)r * (2 * H_) + H_ + c];
    const float hn = g * h[i] + (1.0f - g) * cbuf[i];
    h[i] = hn;
    const int b = r / N_, n = r % N_;
    states[(((long)b * T_ + t) * N_ + n) * H_ + c] = hn;
}

// =====================================================================
extern "C" void kernel_launch(void* const* d_in, const int* in_sizes, int n_in,
                              void* d_out, int out_size, void* d_ws, size_t ws_size,
                              hipStream_t stream)
{
    (void)in_sizes; (void)n_in; (void)out_size; (void)ws_size;
    const float* x    = (const float*)d_in[0];
    const int*   ei   = (const int*)  d_in[1];
    const float* ew   = (const float*)d_in[2];
    const float* Wgmm = (const float*)d_in[3];
    const float* bgmm = (const float*)d_in[4];
    const float* Win  = (const float*)d_in[5];
    const float* bin  = (const float*)d_in[6];
    const float* Wsd  = (const float*)d_in[7];
    const float* bsd  = (const float*)d_in[8];
    const float* Wout = (const float*)d_in[9];
    const float* bout = (const float*)d_in[10];
    const float* Wr   = (const float*)d_in[11];
    const float* br   = (const float*)d_in[12];
    const float* Wu   = (const float*)d_in[13];
    const float* bu   = (const float*)d_in[14];
    const float* Wc   = (const float*)d_in[15];
    const float* bc   = (const float*)d_in[16];

    const int* srcI = ei;
    const int* dstI = ei + E_;

    float* out    = (float*)d_out;
    float* gen    = out;                                     // [B,T,N,161]
    float* pred   = gen  + (size_t)B_ * T_ * N_ * RNNIN_;    // [B,T,N,96]
    float* reps   = pred + (size_t)B_ * T_ * N_ * GMM_;      // [B,T,N,128]
    float* states = reps + (size_t)B_ * T_ * N_ * (2 * H_);  // [1,B,T,N,64]

    // ---- workspace carve-out ----
    float* ws     = (float*)d_ws;
    float* h      = ws;  ws += (size_t)R_ * H_;
    float* s1     = ws;  ws += (size_t)R_ * GMM_;
    float* zin    = ws;  ws += (size_t)R_ * ZLD_;
    float* sdin   = ws;  ws += (size_t)R_ * 2 * H_;    // [z | agg(z)]
    float* conv   = ws;  ws += (size_t)R_ * H_;
    float* reprb  = ws;  ws += (size_t)R_ * 2 * H_;
    float* xhat2  = ws;  ws += (size_t)R_ * XLD_;
    float* diffin = ws;  ws += (size_t)R_ * DLD_;      // [front | a1 | a2]
    float* rg     = ws;  ws += (size_t)R_ * 2 * H_;    // [r | g] fused
    float* cbuf   = ws;  ws += (size_t)R_ * H_;
    // packed weights (Kpad x Npad each)
    float* Wp_gmm = ws;  ws += (size_t)64  * 96;
    float* Wp_in  = ws;  ws += (size_t)164 * 64;
    float* Wp_sd  = ws;  ws += (size_t)128 * 64;
    float* Wp_out = ws;  ws += (size_t)128 * 192;
    float* Wp_ru  = ws;  ws += (size_t)676 * 128;      // [W_r | W_u]
    float* Wp_c   = ws;  ws += (size_t)676 * 64;
    float* b_ru   = ws;  ws += 128;

    auto cdiv = [](long a, long b) -> int { return (int)((a + b - 1) / b); };
    auto pack = [&](const float* W0, int n0, const float* W1, int n1,
                    float* Wp, int K, int Kpad, int Npad) {
        pack_w_dual<<<cdiv((long)Kpad * Npad, 256), 256, 0, stream>>>(W0, n0, W1, n1, Wp, K, Kpad, Npad);
    };
    auto gemm = [&](const float* A, int lda, const float* Wp, int Npad, const float* bias,
                    float* C, int ldc, int Kpad, int Ncols, int act) {
        const int tilesM = R_ / 16, groupsN = Npad / 32;
        gemm_wmma_nt2<<<cdiv((long)tilesM * groupsN, 8), 256, 0, stream>>>(
            A, lda, Wp, Npad, bias, C, ldc, tilesM, groupsN, Kpad, Ncols, act);
    };
    auto agg = [&](const float* in, float* outp, int width, int ld) {
        agg_edges<<<cdiv((long)B_ * E_ * width, 256), 256, 0, stream>>>(
            in, outp, srcI, dstI, ew, width, ld);
    };

    // ---- once per launch: pack weights, zero h and padded activation buffers ----
    pack(Wgmm, GMM_,   nullptr, 0,  Wp_gmm, H_,     64,  96);
    pack(Win,  H_,     nullptr, 0,  Wp_in,  RNNIN_, 164, 64);
    pack(Wsd,  H_,     nullptr, 0,  Wp_sd,  2 * H_, 128, 64);
    pack(Wout, RNNIN_, nullptr, 0,  Wp_out, 2 * H_, 128, 192);
    pack(Wr,   H_,     Wu,      H_, Wp_ru,  DIFF_,  676, 128);
    pack(Wc,   H_,     nullptr, 0,  Wp_c,   DIFF_,  676, 64);
    pack_bias_dual<<<1, 128, 0, stream>>>(br, H_, bu, H_, b_ru);
    fill0<<<cdiv((long)R_ * H_, 256), 256, 0, stream>>>(h, (long)R_ * H_);
    fill0<<<cdiv((long)R_ * ZLD_, 256), 256, 0, stream>>>(zin, (long)R_ * ZLD_);
    fill0<<<cdiv((long)R_ * DLD_, 256), 256, 0, stream>>>(diffin, (long)R_ * DLD_);

    for (int t = 0; t < T_; ++t) {
        // raw = h @ W_gmm + b_gmm
        gemm(h, H_, Wp_gmm, 96, bgmm, s1, GMM_, 64, GMM_, ACT_NONE);
        gmm_head<<<cdiv(R_, 8), 256, 0, stream>>>(s1, pred, t);
        assemble_zin<<<cdiv((long)R_ * RNNIN_, 256), 256, 0, stream>>>(x, s1, h, zin, t);
        // z -> sdin[:,0:64]
        gemm(zin, ZLD_, Wp_in, 64, bin, sdin, 2 * H_, 164, H_, ACT_NONE);
        fill0_2d<<<cdiv((long)R_ * H_, 256), 256, 0, stream>>>(sdin + H_, R_, H_, 2 * H_);
        agg(sdin, sdin + H_, H_, 2 * H_);
        gemm(sdin, 2 * H_, Wp_sd, 64, bsd, conv, H_, 128, H_, ACT_NONE);
        assemble_repr<<<cdiv((long)R_ * 2 * H_, 256), 256, 0, stream>>>(conv, h, reprb, reps, t);
        gemm(reprb, 2 * H_, Wp_out, 192, bout, xhat2, XLD_, 128, RNNIN_, ACT_NONE);
        copy_gen<<<cdiv((long)R_ * RNNIN_, 256), 256, 0, stream>>>(xhat2, gen, t);
        // diffin = [xh | agg(xh) | agg(agg(xh))] shared by r and g
        assemble_front<<<cdiv((long)R_ * FRONT_, 256), 256, 0, stream>>>(xhat2, h, rg, diffin, 0);
        fill0_2d<<<cdiv((long)R_ * 2 * FRONT_, 256), 256, 0, stream>>>(diffin + FRONT_, R_, 2 * FRONT_, DLD_);
        agg(diffin, diffin + FRONT_, FRONT_, DLD_);
        agg(diffin + FRONT_, diffin + 2 * FRONT_, FRONT_, DLD_);
        // fused [r|g] = sigmoid(diffin @ [W_r|W_u] + [b_r|b_u])
        gemm(diffin, DLD_, Wp_ru, 128, b_ru, rg, 2 * H_, 676, 2 * H_, ACT_SIGMOID);
        // diffin = [cin | agg | agg^2], cin = [xs_hat_2 | r*h]
        assemble_front<<<cdiv((long)R_ * FRONT_, 256), 256, 0, stream>>>(xhat2, h, rg, diffin, 1);
        fill0_2d<<<cdiv((long)R_ * 2 * FRONT_, 256), 256, 0, stream>>>(diffin + FRONT_, R_, 2 * FRONT_, DLD_);
        agg(diffin, diffin + FRONT_, FRONT_, DLD_);
        agg(diffin + FRONT_, diffin + 2 * FRONT_, FRONT_, DLD_);
        gemm(diffin, DLD_, Wp_c, 64, bc, cbuf, H_, 676, H_, ACT_TANH);
        h_update<<<cdiv((long)R_ * H_, 256), 256, 0, stream>>>(h, rg, cbuf, states, t);
    }
}